// TransformerConv_56607668961465
// MI455X (gfx1250) — compile-verified
//
#include <hip/hip_runtime.h>
#include <math.h>

// ---------------------------------------------------------------------------
// Problem constants (from the reference)
// ---------------------------------------------------------------------------
#define NN 50000
#define EE 800000
#define CC 64
#define AA 4
#define BB 16
#define HH 8
#define DD 32
#define PP 16

typedef float v2f __attribute__((ext_vector_type(2)));
typedef float v8f __attribute__((ext_vector_type(8)));

__device__ __forceinline__ v8f vzero8() {
    v8f r = {0.f, 0.f, 0.f, 0.f, 0.f, 0.f, 0.f, 0.f};
    return r;
}

// fp32 WMMA 16x16x4: D = A(16x4) * B(4x16) + C(16x16), fp32 accumulate.
// A layout: lane (l&15)=row M, VGPR pair holds K = 2*(l>>4), 2*(l>>4)+1
// C/D layout: v8f elem i -> row (i + 8*(l>>4)), col (l&15)
__device__ __forceinline__ v8f wmma4(v2f a, v2f b, v8f c) {
    return __builtin_amdgcn_wmma_f32_16x16x4_f32(
        /*neg_a=*/false, a, /*neg_b=*/false, b,
        /*c_mod=*/(short)0, c, /*reuse_a=*/false, /*reuse_b=*/false);
}

// Weight matrices are staged in LDS "K-pair interleaved": element (k,n) lives
// at ((k>>1)*Ncol + n)*2 + (k&1).  A WMMA B-fragment (rows k0,k0+1, col n,
// k0 even) is then one aligned 8-byte ds load:  *(v2f*)&sW[(kp*Ncol + n)*2].

// ---------------------------------------------------------------------------
// Kernel 0: zero the softmax normalizer
// ---------------------------------------------------------------------------
__global__ void zero_f32_kernel(float* __restrict__ p, int n) {
    int i = blockIdx.x * blockDim.x + threadIdx.x;
    if (i < n) p[i] = 0.f;
}

// ---------------------------------------------------------------------------
// Kernel 1: per-node precompute.
//   q_node = node_feats @ W_q                       [N, 32]
//   sc     = einsum('nc,np,cpo->no', f, a, W_sc)    [N, 64]  -> written to out
// One wave = 16 nodes; 128 threads = 4 waves per block.
// ---------------------------------------------------------------------------
__global__ void __launch_bounds__(128)
node_pre_kernel(const float* __restrict__ feats, const float* __restrict__ attrs,
                const float* __restrict__ Wq, const float* __restrict__ Wsc,
                float* __restrict__ q_node, float* __restrict__ sc_out) {
    __shared__ __attribute__((aligned(16))) float sWq[CC * DD];      // interleaved
    __shared__ __attribute__((aligned(16))) float sWsc[64 * CC];     // interleaved chunk
    __shared__ __attribute__((aligned(16))) float sF[4][16 * CC];
    __shared__ __attribute__((aligned(16))) float sP[4][16 * PP];

    const int tid = threadIdx.x;
    const int wave = tid >> 5, lane = tid & 31;
    const int lo = lane & 15, hi = lane >> 4;

    int tile = blockIdx.x * 4 + wave;
    const int ntiles = NN / 16;                 // 3125 exact
    if (tile >= ntiles) tile = ntiles - 1;      // duplicate tail work (benign)
    const int nbase = tile * 16;

    for (int i = tid; i < CC * DD; i += 128) {
        int k = i >> 5, n = i & 31;
        sWq[((k >> 1) * DD + n) * 2 + (k & 1)] = Wq[i];
    }

    for (int it = 0; it < 8; ++it) {            // 16 rows x 16 float4
        int idx = lane + 32 * it; int r = idx >> 4, q4 = idx & 15;
        *(float4*)&sF[wave][r * CC + q4 * 4] =
            *(const float4*)&feats[(long)(nbase + r) * CC + q4 * 4];
    }
    for (int it = 0; it < 2; ++it) {            // 16 rows x 4 float4
        int idx = lane + 32 * it; int r = idx >> 2, q4 = idx & 3;
        *(float4*)&sP[wave][r * PP + q4 * 4] =
            *(const float4*)&attrs[(long)(nbase + r) * PP + q4 * 4];
    }
    __syncthreads();

    // ---- q GEMM: [16,64] x [64,32] ----
    {
        v8f acc[2]; acc[0] = vzero8(); acc[1] = vzero8();
        for (int ks = 0; ks < 16; ++ks) {
            int k0 = ks * 4 + 2 * hi;
            int kp = ks * 2 + hi;
            v2f a = *(const v2f*)&sF[wave][lo * CC + k0];
            #pragma unroll
            for (int nt = 0; nt < 2; ++nt) {
                v2f b = *(const v2f*)&sWq[(kp * DD + nt * 16 + lo) * 2];
                acc[nt] = wmma4(a, b, acc[nt]);
            }
        }
        for (int nt = 0; nt < 2; ++nt)
            for (int i = 0; i < 8; ++i) {
                int r = i + 8 * hi;
                q_node[(long)(nbase + r) * DD + nt * 16 + lo] = acc[nt][i];
            }
    }

    // ---- self-connection GEMM: fa[16,1024] @ Wsc[1024,64] ----
    v8f accs[4];
    for (int t = 0; t < 4; ++t) accs[t] = vzero8();
    for (int ch = 0; ch < 16; ++ch) {
        __syncthreads();
        for (int i = tid; i < 64 * CC; i += 128) {
            int k = i >> 6, n = i & 63;
            sWsc[((k >> 1) * CC + n) * 2 + (k & 1)] = Wsc[(long)ch * 64 * CC + i];
        }
        __syncthreads();
        for (int ks = 0; ks < 16; ++ks) {
            int kl = ks * 4 + 2 * hi;           // local K (even)
            int kk = ch * 64 + kl;              // global K = c*16 + p
            int kp = ks * 2 + hi;               // local K-pair
            float fv = sF[wave][lo * CC + (kk >> 4)];           // same c for kk,kk+1
            v2f pv = *(const v2f*)&sP[wave][lo * PP + (kk & 15)];
            v2f a; a.x = fv * pv.x; a.y = fv * pv.y;
            #pragma unroll
            for (int nt = 0; nt < 4; ++nt) {
                v2f b = *(const v2f*)&sWsc[(kp * CC + nt * 16 + lo) * 2];
                accs[nt] = wmma4(a, b, accs[nt]);
            }
        }
    }
    for (int nt = 0; nt < 4; ++nt)
        for (int i = 0; i < 8; ++i) {
            int r = i + 8 * hi;
            sc_out[(long)(nbase + r) * CC + nt * 16 + lo] = accs[nt][i];
        }
}

// ---------------------------------------------------------------------------
// Kernel 2: edge score pass.  One wave = 16 edges.  64 threads = 2 waves.
// ---------------------------------------------------------------------------
__global__ void __launch_bounds__(64)
edge_score_kernel(const float* __restrict__ feats, const float* __restrict__ emb,
                  const float* __restrict__ eattr, const float* __restrict__ pos,
                  const int* __restrict__ esrc, const int* __restrict__ edst,
                  const float* __restrict__ W1, const float* __restrict__ W2,
                  const float* __restrict__ Wl, const float* __restrict__ Wdot,
                  const float* __restrict__ q_node,
                  float* __restrict__ expv_buf, float* __restrict__ z) {
    __shared__ __attribute__((aligned(16))) float sW1[BB * HH];
    __shared__ __attribute__((aligned(16))) float sW2[HH * CC * AA];  // interleaved
    __shared__ __attribute__((aligned(16))) float sWl[CC * DD];       // interleaved
    __shared__ __attribute__((aligned(16))) float sWdT[DD * DD];      // transposed+interleaved
    __shared__ __attribute__((aligned(16))) float sEmb[2][16 * BB];
    __shared__ __attribute__((aligned(16))) float sH[2][16 * HH];
    __shared__ __attribute__((aligned(16))) float sG[2][16 * CC];
    __shared__ __attribute__((aligned(16))) float sX[2][16 * CC];
    __shared__ __attribute__((aligned(16))) float sK[2][16 * DD];
    __shared__ __attribute__((aligned(16))) float sQ[2][16 * DD];
    __shared__ __attribute__((aligned(16))) float sAttr[2][16 * AA];
    __shared__ float sScore[2][16];
    __shared__ int sSrc[2][16], sDst[2][16];

    const int tid = threadIdx.x;
    const int wave = tid >> 5, lane = tid & 31;
    const int lo = lane & 15, hi = lane >> 4;
    const int tile = blockIdx.x * 2 + wave;     // E/16 = 50000 exact
    const long ebase = (long)tile * 16;

    __builtin_prefetch(&emb[(ebase + 16) * BB], 0, 1);
    __builtin_prefetch(&eattr[(ebase + 16) * AA], 0, 1);

    for (int i = tid; i < BB * HH; i += 64) sW1[i] = W1[i];
    for (int i = tid; i < HH * 256; i += 64) {
        int k = i >> 8, n = i & 255;
        sW2[((k >> 1) * 256 + n) * 2 + (k & 1)] = W2[i];
    }
    for (int i = tid; i < CC * DD; i += 64) {
        int k = i >> 5, n = i & 31;
        sWl[((k >> 1) * DD + n) * 2 + (k & 1)] = Wl[i];
    }
    for (int i = tid; i < DD * DD; i += 64) {
        int n = i >> 5, k = i & 31;             // B[k][n] = Wdot[n][k]
        sWdT[((k >> 1) * DD + n) * 2 + (k & 1)] = Wdot[i];
    }
    if (lane < 16) {
        sSrc[wave][lane] = esrc[ebase + lane];
        sDst[wave][lane] = edst[ebase + lane];
        *(float4*)&sAttr[wave][lane * 4] = *(const float4*)&eattr[ebase * AA + lane * 4];
    }
    for (int it = 0; it < 2; ++it) {
        int id = lane + 32 * it;
        *(float4*)&sEmb[wave][id * 4] = *(const float4*)&emb[ebase * BB + id * 4];
    }
    __syncthreads();

    for (int it = 0; it < 8; ++it) {
        int idx = lane + 32 * it; int r = idx >> 4, q4 = idx & 15;
        *(float4*)&sX[wave][r * CC + q4 * 4] =
            *(const float4*)&feats[(long)sSrc[wave][r] * CC + q4 * 4];
    }
    for (int it = 0; it < 4; ++it) {
        int idx = lane + 32 * it; int r = idx >> 3, q4 = idx & 7;
        *(float4*)&sQ[wave][r * DD + q4 * 4] =
            *(const float4*)&q_node[(long)sDst[wave][r] * DD + q4 * 4];
    }

    // ---- MLP1 + silu ----
    for (int t = 0; t < 4; ++t) {
        int o = lane + 32 * t; int e = o >> 3, j = o & 7;
        float s = 0.f;
        for (int b = 0; b < BB; ++b) s += sEmb[wave][e * BB + b] * sW1[b * HH + j];
        sH[wave][e * HH + j] = s / (1.f + __expf(-s));
    }
    __syncthreads();

    // ---- MLP2 via WMMA + attr contraction (4-lane butterfly) -> g ----
    {
        float attr_r[8];
        #pragma unroll
        for (int i = 0; i < 8; ++i)
            attr_r[i] = sAttr[wave][(i + 8 * hi) * 4 + (lo & 3)];
        v2f a0 = *(const v2f*)&sH[wave][lo * HH + 2 * hi];
        v2f a1 = *(const v2f*)&sH[wave][lo * HH + 4 + 2 * hi];
        for (int nt = 0; nt < 16; ++nt) {
            v8f acc = vzero8();
            v2f b0 = *(const v2f*)&sW2[(hi * 256 + nt * 16 + lo) * 2];
            v2f b1 = *(const v2f*)&sW2[((2 + hi) * 256 + nt * 16 + lo) * 2];
            acc = wmma4(a0, b0, acc);
            acc = wmma4(a1, b1, acc);
            #pragma unroll
            for (int i = 0; i < 8; ++i) {
                float t = acc[i] * attr_r[i];
                t += __shfl_xor(t, 1, 32);
                t += __shfl_xor(t, 2, 32);
                // 4 lanes of each group hold the same g; duplicate store is benign
                sG[wave][(i + 8 * hi) * CC + nt * 4 + (lo >> 2)] = 0.5f * t;
            }
        }
    }
    __syncthreads();

    // ---- k = (x * g) @ W_lin_k : [16,64] x [64,32] ----
    {
        v8f acc[2]; acc[0] = vzero8(); acc[1] = vzero8();
        for (int ks = 0; ks < 16; ++ks) {
            int k0 = ks * 4 + 2 * hi;
            int kp = ks * 2 + hi;
            v2f xv = *(const v2f*)&sX[wave][lo * CC + k0];
            v2f gv = *(const v2f*)&sG[wave][lo * CC + k0];
            v2f a = xv * gv;
            #pragma unroll
            for (int nt = 0; nt < 2; ++nt) {
                v2f b = *(const v2f*)&sWl[(kp * DD + nt * 16 + lo) * 2];
                acc[nt] = wmma4(a, b, acc[nt]);
            }
        }
        for (int nt = 0; nt < 2; ++nt)
            for (int i = 0; i < 8; ++i)
                sK[wave][(i + 8 * hi) * DD + nt * 16 + lo] = acc[nt][i];
    }
    __syncthreads();

    // ---- kd = k @ W_dot^T, then score = rowwise dot with q (lane butterfly) ----
    {
        v8f acc[2]; acc[0] = vzero8(); acc[1] = vzero8();
        for (int ks = 0; ks < 8; ++ks) {
            int k0 = ks * 4 + 2 * hi;
            int kp = ks * 2 + hi;
            v2f a = *(const v2f*)&sK[wave][lo * DD + k0];
            #pragma unroll
            for (int nt = 0; nt < 2; ++nt) {
                v2f b = *(const v2f*)&sWdT[(kp * DD + nt * 16 + lo) * 2];
                acc[nt] = wmma4(a, b, acc[nt]);
            }
        }
        #pragma unroll
        for (int i = 0; i < 8; ++i) {
            int r = i + 8 * hi;
            float t = acc[0][i] * sQ[wave][r * DD + lo]
                    + acc[1][i] * sQ[wave][r * DD + 16 + lo];
            t += __shfl_xor(t, 1, 32);
            t += __shfl_xor(t, 2, 32);
            t += __shfl_xor(t, 4, 32);
            t += __shfl_xor(t, 8, 32);
            sScore[wave][r] = t;        // 16 lanes store identical value
        }
    }
    __syncthreads();

    // ---- per-edge cutoff / expv (lanes 0..15) ----
    if (lane < 16) {
        int e = lane;
        float sc = sScore[wave][e];
        int s = sSrc[wave][e], d = sDst[wave][e];
        float dx = pos[(long)d * 3 + 0] - pos[(long)s * 3 + 0];
        float dy = pos[(long)d * 3 + 1] - pos[(long)s * 3 + 1];
        float dz = pos[(long)d * 3 + 2] - pos[(long)s * 3 + 2];
        float len = sqrtf(dx * dx + dy * dy + dz * dz);
        float xarg = 10.f - 2.f * len;          // 2*R_MAX*(1 - len/R_MAX)
        float cut = (xarg > 0.f) ? __expf(-1.f / xarg) : 0.f;
        float ev = cut * __expf(sc);
        expv_buf[ebase + e] = ev;
        atomicAdd(&z[d], ev);
    }
}

// ---------------------------------------------------------------------------
// Kernel 3: edge value pass.  One wave = 16 edges.  64 threads = 2 waves.
// ---------------------------------------------------------------------------
__global__ void __launch_bounds__(64)
edge_value_kernel(const float* __restrict__ feats, const float* __restrict__ emb,
                  const float* __restrict__ eattr,
                  const int* __restrict__ esrc, const int* __restrict__ edst,
                  const float* __restrict__ W1, const float* __restrict__ W2,
                  const float* __restrict__ Wl,
                  const float* __restrict__ expv_buf, const float* __restrict__ z,
                  float* __restrict__ out) {
    __shared__ __attribute__((aligned(16))) float sW1[BB * HH];
    __shared__ __attribute__((aligned(16))) float sW2[HH * CC * AA];  // interleaved
    __shared__ __attribute__((aligned(16))) float sWl[CC * CC];       // interleaved
    __shared__ __attribute__((aligned(16))) float sEmb[2][16 * BB];
    __shared__ __attribute__((aligned(16))) float sH[2][16 * HH];
    __shared__ __attribute__((aligned(16))) float sG[2][16 * CC];
    __shared__ __attribute__((aligned(16))) float sX[2][16 * CC];
    __shared__ __attribute__((aligned(16))) float sAttr[2][16 * AA];
    __shared__ float sM[2][16];
    __shared__ int sSrc[2][16], sDst[2][16];

    const int tid = threadIdx.x;
    const int wave = tid >> 5, lane = tid & 31;
    const int lo = lane & 15, hi = lane >> 4;
    const int tile = blockIdx.x * 2 + wave;
    const long ebase = (long)tile * 16;

    __builtin_prefetch(&emb[(ebase + 16) * BB], 0, 1);
    __builtin_prefetch(&eattr[(ebase + 16) * AA], 0, 1);

    for (int i = tid; i < BB * HH; i += 64) sW1[i] = W1[i];
    for (int i = tid; i < HH * 256; i += 64) {
        int k = i >> 8, n = i & 255;
        sW2[((k >> 1) * 256 + n) * 2 + (k & 1)] = W2[i];
    }
    for (int i = tid; i < CC * CC; i += 64) {
        int k = i >> 6, n = i & 63;
        sWl[((k >> 1) * CC + n) * 2 + (k & 1)] = Wl[i];
    }
    if (lane < 16) {
        sSrc[wave][lane] = esrc[ebase + lane];
        sDst[wave][lane] = edst[ebase + lane];
        *(float4*)&sAttr[wave][lane * 4] = *(const float4*)&eattr[ebase * AA + lane * 4];
    }
    for (int it = 0; it < 2; ++it) {
        int id = lane + 32 * it;
        *(float4*)&sEmb[wave][id * 4] = *(const float4*)&emb[ebase * BB + id * 4];
    }
    __syncthreads();

    for (int it = 0; it < 8; ++it) {
        int idx = lane + 32 * it; int r = idx >> 4, q4 = idx & 15;
        *(float4*)&sX[wave][r * CC + q4 * 4] =
            *(const float4*)&feats[(long)sSrc[wave][r] * CC + q4 * 4];
    }

    // softmax scale per edge
    if (lane < 16) {
        float zz = z[sDst[wave][lane]];
        zz = (zz == 0.f) ? 1.f : zz;
        float al = expv_buf[ebase + lane] / zz;
        sM[wave][lane] = sqrtf(fmaxf(al, 0.f));
    }

    // ---- MLP1 + silu ----
    for (int t = 0; t < 4; ++t) {
        int o = lane + 32 * t; int e = o >> 3, j = o & 7;
        float s = 0.f;
        for (int b = 0; b < BB; ++b) s += sEmb[wave][e * BB + b] * sW1[b * HH + j];
        sH[wave][e * HH + j] = s / (1.f + __expf(-s));
    }
    __syncthreads();

    // ---- MLP2 via WMMA + attr contraction (butterfly) -> g_v ----
    {
        float attr_r[8];
        #pragma unroll
        for (int i = 0; i < 8; ++i)
            attr_r[i] = sAttr[wave][(i + 8 * hi) * 4 + (lo & 3)];
        v2f a0 = *(const v2f*)&sH[wave][lo * HH + 2 * hi];
        v2f a1 = *(const v2f*)&sH[wave][lo * HH + 4 + 2 * hi];
        for (int nt = 0; nt < 16; ++nt) {
            v8f acc = vzero8();
            v2f b0 = *(const v2f*)&sW2[(hi * 256 + nt * 16 + lo) * 2];
            v2f b1 = *(const v2f*)&sW2[((2 + hi) * 256 + nt * 16 + lo) * 2];
            acc = wmma4(a0, b0, acc);
            acc = wmma4(a1, b1, acc);
            #pragma unroll
            for (int i = 0; i < 8; ++i) {
                float t = acc[i] * attr_r[i];
                t += __shfl_xor(t, 1, 32);
                t += __shfl_xor(t, 2, 32);
                sG[wave][(i + 8 * hi) * CC + nt * 4 + (lo >> 2)] = 0.5f * t;
            }
        }
    }
    __syncthreads();

    // ---- v = (x * g) @ W_lin_v : [16,64] x [64,64]; scale + scatter ----
    {
        v8f acc[4];
        for (int t = 0; t < 4; ++t) acc[t] = vzero8();
        for (int ks = 0; ks < 16; ++ks) {
            int k0 = ks * 4 + 2 * hi;
            int kp = ks * 2 + hi;
            v2f xv = *(const v2f*)&sX[wave][lo * CC + k0];
            v2f gv = *(const v2f*)&sG[wave][lo * CC + k0];
            v2f a = xv * gv;
            #pragma unroll
            for (int nt = 0; nt < 4; ++nt) {
                v2f b = *(const v2f*)&sWl[(kp * CC + nt * 16 + lo) * 2];
                acc[nt] = wmma4(a, b, acc[nt]);
            }
        }
        for (int nt = 0; nt < 4; ++nt)
            for (int i = 0; i < 8; ++i) {
                int r = i + 8 * hi;
                float val = acc[nt][i] * sM[wave][r];
                atomicAdd(&out[(long)sDst[wave][r] * CC + nt * 16 + lo], val);
            }
    }
}

// ---------------------------------------------------------------------------
// Host-side launch
// ---------------------------------------------------------------------------
extern "C" void kernel_launch(void* const* d_in, const int* in_sizes, int n_in,
                              void* d_out, int out_size, void* d_ws, size_t ws_size,
                              hipStream_t stream) {
    const float* node_feats = (const float*)d_in[0];
    const float* node_attrs = (const float*)d_in[1];
    const float* edge_emb   = (const float*)d_in[2];
    const float* edge_attrs = (const float*)d_in[3];
    const float* positions  = (const float*)d_in[4];
    const int*   edge_src   = (const int*)d_in[5];
    const int*   edge_dst   = (const int*)d_in[6];
    const float* W_q     = (const float*)d_in[7];
    const float* W1_k    = (const float*)d_in[8];
    const float* W2_k    = (const float*)d_in[9];
    const float* W1_v    = (const float*)d_in[10];
    const float* W2_v    = (const float*)d_in[11];
    const float* W_lin_k = (const float*)d_in[12];
    const float* W_lin_v = (const float*)d_in[13];
    const float* W_dot   = (const float*)d_in[14];
    const float* W_sc    = (const float*)d_in[15];
    float* out = (float*)d_out;

    // workspace layout (floats): q_node[N*32] | z[N] | expv[E]  (~10 MB)
    float* q_node = (float*)d_ws;
    float* z      = q_node + (size_t)NN * DD;
    float* expv   = z + NN;

    zero_f32_kernel<<<(NN + 255) / 256, 256, 0, stream>>>(z, NN);

    // 3125 node tiles, 4 waves/block of 128 threads
    node_pre_kernel<<<(NN / 16 + 3) / 4, 128, 0, stream>>>(
        node_feats, node_attrs, W_q, W_sc, q_node, out);

    // 50000 edge tiles, 2 waves/block of 64 threads
    edge_score_kernel<<<EE / 16 / 2, 64, 0, stream>>>(
        node_feats, edge_emb, edge_attrs, positions, edge_src, edge_dst,
        W1_k, W2_k, W_lin_k, W_dot, q_node, expv, z);

    edge_value_kernel<<<EE / 16 / 2, 64, 0, stream>>>(
        node_feats, edge_emb, edge_attrs, edge_src, edge_dst,
        W1_v, W2_v, W_lin_v, expv, z, out);
}